// GCNII_withoutLinear_91096256348434
// MI455X (gfx1250) — compile-verified
//
#include <hip/hip_runtime.h>
#include <math.h>

// ---------------------------------------------------------------------------
// GCNII forward on gfx1250 (MI455X). fp32 WMMA (V_WMMA_F32_16X16X4_F32) for
// all dense GEMMs; wave-per-edge f32 atomic scatter-add SpMM (h fits in L2).
// hh-initialization (ALPHA*x0) is fused into the GEMM epilogues.
// ---------------------------------------------------------------------------

typedef __attribute__((ext_vector_type(2))) float v2f;
typedef __attribute__((ext_vector_type(8))) float v8f;

__device__ __forceinline__ v8f wmma4(v2f a, v2f b, v8f c) {
  // (neg_a, A, neg_b, B, c_mod, C, reuse_a, reuse_b)
  return __builtin_amdgcn_wmma_f32_16x16x4_f32(false, a, false, b,
                                               (short)0, c, false, false);
}

// ---------------------------------------------------------------------------
// h = relu(x @ W_in + b_in); also writes x0 = h and hh = ALPHA*h (layer-0
// residual init, so no separate init kernel is needed).
// A: [M,128], B: [128,64]. One block = 16 rows, 4 waves cover N=64.
// ---------------------------------------------------------------------------
__global__ void __launch_bounds__(128)
gemm_in_kernel(const float* __restrict__ A, const float* __restrict__ B,
               const float* __restrict__ bias, float* __restrict__ h,
               float* __restrict__ x0, float* __restrict__ hh,
               float alpha, int M) {
  constexpr int K = 128, N = 64, LDA = K + 1;
  __shared__ float As[16 * LDA];
  const int m0 = blockIdx.x * 16;

  for (int i = threadIdx.x; i < 16 * K; i += 128) {
    const int r = i >> 7, c = i & 127;
    As[r * LDA + c] = A[(size_t)(m0 + r) * K + c];
  }
  __syncthreads();

  const int lane = threadIdx.x & 31;
  const int wave = threadIdx.x >> 5;      // 0..3 -> n-tile
  const int n0   = wave * 16;
  const int t    = lane & 15;
  const int kh   = (lane >> 4) << 1;      // 0 or 2

  v8f acc = {};
  for (int k0 = 0; k0 < K; k0 += 4) {
    v2f a, b;
    a.x = As[t * LDA + k0 + kh];
    a.y = As[t * LDA + k0 + kh + 1];
    b.x = B[(size_t)(k0 + kh) * N + n0 + t];
    b.y = B[(size_t)(k0 + kh + 1) * N + n0 + t];
    acc = wmma4(a, b, acc);
  }

  const int col   = n0 + t;
  const int rbase = m0 + ((lane >> 4) << 3);
  const float bv  = bias[col];
#pragma unroll
  for (int r = 0; r < 8; ++r) {
    float v = acc[r] + bv;
    v = v > 0.f ? v : 0.f;
    const size_t o = (size_t)(rbase + r) * N + col;
    h[o]  = v;
    x0[o] = v;
    hh[o] = alpha * v;                    // residual-mix base for layer 0 SpMM
  }
}

// ---------------------------------------------------------------------------
// hh[dst] += (1-ALPHA) * w[e] * h[src].  One wave32 per edge: edge metadata is
// wave-uniform (readfirstlane -> scalar loads); each lane moves a float2, so a
// wave covers one full 256B feature row with coalesced b64 loads + 2 atomics.
// ---------------------------------------------------------------------------
__global__ void __launch_bounds__(256)
spmm_kernel(const float* __restrict__ h, const int* __restrict__ src,
            const int* __restrict__ dst, const float* __restrict__ w,
            float* __restrict__ hh, int E, float scale) {
  const int wid = blockIdx.x * (blockDim.x >> 5) + (threadIdx.x >> 5);
  if (wid >= E) return;                    // wave-uniform branch
  const int e    = __builtin_amdgcn_readfirstlane(wid);
  const int lane = threadIdx.x & 31;
  const int   s  = src[e];
  const int   d  = dst[e];
  const float we = scale * w[e];
  const float2 hv = ((const float2*)h)[(size_t)s * 32 + lane];
  float* dp = hh + (size_t)d * 64 + (lane << 1);
  atomicAdd(dp,     we * hv.x);
  atomicAdd(dp + 1, we * hv.y);
}

// ---------------------------------------------------------------------------
// h = relu((1-beta)*hh + beta*(hh @ W))   A=hh: [M,64], W: [64,64]
// hh tile reused from LDS for the identity-mapping epilogue; then this
// block's hh rows are re-initialized to ALPHA*x0 for the next layer
// (safe: row-tiles of hh are block-exclusive and fully staged in LDS).
// ---------------------------------------------------------------------------
__global__ void __launch_bounds__(128)
gemm_conv_kernel(const float* __restrict__ HH, const float* __restrict__ W,
                 float* __restrict__ Hout, const float* __restrict__ x0,
                 float beta, float alpha, int init_next, int M) {
  constexpr int K = 64, N = 64, LDA = K + 1;
  __shared__ float As[16 * LDA];
  const int m0 = blockIdx.x * 16;

  for (int i = threadIdx.x; i < 16 * K; i += 128) {
    const int r = i >> 6, c = i & 63;
    As[r * LDA + c] = HH[(size_t)(m0 + r) * K + c];
  }
  __syncthreads();

  const int lane = threadIdx.x & 31;
  const int wave = threadIdx.x >> 5;
  const int n0   = wave * 16;
  const int t    = lane & 15;
  const int kh   = (lane >> 4) << 1;

  v8f acc = {};
  for (int k0 = 0; k0 < K; k0 += 4) {
    v2f a, b;
    a.x = As[t * LDA + k0 + kh];
    a.y = As[t * LDA + k0 + kh + 1];
    b.x = W[(size_t)(k0 + kh) * N + n0 + t];
    b.y = W[(size_t)(k0 + kh + 1) * N + n0 + t];
    acc = wmma4(a, b, acc);
  }

  const int col  = n0 + t;
  const int rloc = (lane >> 4) << 3;      // row offset within tile
  const float ob = 1.f - beta;
  float* hh_mut = const_cast<float*>(HH);
#pragma unroll
  for (int r = 0; r < 8; ++r) {
    const float hhv = As[(rloc + r) * LDA + col];  // K==N==64: reuse LDS tile
    float v = ob * hhv + beta * acc[r];
    v = v > 0.f ? v : 0.f;
    const size_t o = (size_t)(m0 + rloc + r) * N + col;
    Hout[o] = v;
    if (init_next) hh_mut[o] = alpha * x0[o];      // prep next layer's SpMM
  }
}

// ---------------------------------------------------------------------------
// logits = h @ W_out + b_out   A: [M,64], W: [64,40] (3 n-tiles, masked)
// ---------------------------------------------------------------------------
__global__ void __launch_bounds__(96)
gemm_out_kernel(const float* __restrict__ H, const float* __restrict__ W,
                const float* __restrict__ bias, float* __restrict__ out,
                int M) {
  constexpr int K = 64, N = 40, LDA = K + 1;
  __shared__ float As[16 * LDA];
  const int m0 = blockIdx.x * 16;

  for (int i = threadIdx.x; i < 16 * K; i += 96) {
    const int r = i >> 6, c = i & 63;
    As[r * LDA + c] = H[(size_t)(m0 + r) * K + c];
  }
  __syncthreads();

  const int lane = threadIdx.x & 31;
  const int wave = threadIdx.x >> 5;      // 0..2
  const int n0   = wave * 16;
  const int t    = lane & 15;
  const int kh   = (lane >> 4) << 1;
  const int nn   = n0 + t;
  const bool nok = (nn < N);

  v8f acc = {};
  for (int k0 = 0; k0 < K; k0 += 4) {
    v2f a, b;
    a.x = As[t * LDA + k0 + kh];
    a.y = As[t * LDA + k0 + kh + 1];
    b.x = nok ? W[(size_t)(k0 + kh) * N + nn] : 0.f;
    b.y = nok ? W[(size_t)(k0 + kh + 1) * N + nn] : 0.f;
    acc = wmma4(a, b, acc);
  }

  if (nok) {
    const int rbase = m0 + ((lane >> 4) << 3);
    const float bv = bias[nn];
#pragma unroll
    for (int r = 0; r < 8; ++r)
      out[(size_t)(rbase + r) * N + nn] = acc[r] + bv;
  }
}

// log_softmax over 40 classes; one wave32 per row (lanes cover f and f+32).
__global__ void log_softmax_kernel(float* __restrict__ out, int M, int C) {
  const int row = blockIdx.x * (blockDim.x >> 5) + (threadIdx.x >> 5);
  if (row >= M) return;
  const int lane = threadIdx.x & 31;
  const float NEG = -3.402823466e38f;
  float v0 = (lane < C)      ? out[(size_t)row * C + lane]      : NEG;
  float v1 = (lane + 32 < C) ? out[(size_t)row * C + lane + 32] : NEG;
  float m = fmaxf(v0, v1);
  for (int off = 16; off > 0; off >>= 1) m = fmaxf(m, __shfl_xor(m, off, 32));
  float s = 0.f;
  if (lane < C)      s += expf(v0 - m);
  if (lane + 32 < C) s += expf(v1 - m);
  for (int off = 16; off > 0; off >>= 1) s += __shfl_xor(s, off, 32);
  const float lse = m + logf(s);
  if (lane < C)      out[(size_t)row * C + lane]      = v0 - lse;
  if (lane + 32 < C) out[(size_t)row * C + lane + 32] = v1 - lse;
}

// ---------------------------------------------------------------------------
extern "C" void kernel_launch(void* const* d_in, const int* in_sizes, int n_in,
                              void* d_out, int out_size, void* d_ws, size_t ws_size,
                              hipStream_t stream) {
  const float* x     = (const float*)d_in[0];
  const int*   esrc  = (const int*)d_in[1];
  const int*   edst  = (const int*)d_in[2];
  const float* ew    = (const float*)d_in[3];
  const float* W_in  = (const float*)d_in[4];
  const float* b_in  = (const float*)d_in[5];
  const float* convW = (const float*)d_in[6];
  const float* W_out = (const float*)d_in[7];
  const float* b_out = (const float*)d_in[8];
  float* out = (float*)d_out;

  const int F_IN = 128, HID = 64, N_CLS = 40;
  const float ALPHA = 0.1f, THETA = 0.5f;
  const int M = in_sizes[0] / F_IN;          // 170000 (divisible by 16)
  const int E = in_sizes[1];                 // 1200000
  const int n_convs = in_sizes[6] / (HID * HID);

  const size_t sz = (size_t)M * HID;
  float* x0 = (float*)d_ws;                  // [M,64]
  float* h  = x0 + sz;                       // [M,64]
  float* hh = h + sz;                        // [M,64]

  const int mtiles = (M + 15) / 16;
  const int spmm_blocks = (E * 32 + 255) / 256;   // one wave32 per edge

  gemm_in_kernel<<<mtiles, 128, 0, stream>>>(x, W_in, b_in, h, x0, hh, ALPHA, M);

  for (int l = 0; l < n_convs; ++l) {
    const float beta = logf(THETA / (float)(l + 1) + 1.0f);
    spmm_kernel<<<spmm_blocks, 256, 0, stream>>>(h, esrc, edst, ew, hh,
                                                 E, 1.0f - ALPHA);
    gemm_conv_kernel<<<mtiles, 128, 0, stream>>>(hh, convW + (size_t)l * HID * HID,
                                                 h, x0, beta, ALPHA,
                                                 (l + 1 < n_convs) ? 1 : 0, M);
  }

  gemm_out_kernel<<<mtiles, 96, 0, stream>>>(h, W_out, b_out, out, M);
  log_softmax_kernel<<<(M + 7) / 8, 256, 0, stream>>>(out, M, N_CLS);
}